// MultiHeadAttentionBlock_4587025072621
// MI455X (gfx1250) — compile-verified
//
#include <hip/hip_runtime.h>

// ---------------------------------------------------------------------------
// MultiHeadAttentionBlock, algebraically reduced (see analysis):
//   softmax rows sum to 1  =>  attention output == broadcast of
//   (Sum_s v[n,s,:]) @ Wv^T * 1 ... folded as:
//     vsum[n,e]  = sum_s v[n,s,e]
//     Vsum[n,f]  = sum_e vsum[n,e]*Wv[f,e] + 2048*bv[f]
//     O[n,f]     = sum_e Vsum[n,e]*Wo[f,e] + bo[f]
//     out[n,s,f] = O[n,f]   for all s
// q, k, mask, Wq, bq, Wk, bk do not affect the output.
// Bandwidth-bound: ~40 MB traffic -> ~1.7 us at 23.3 TB/s.
// The two 2x1024x1024 projections run on V_WMMA_F32_16X16X4_F32 (full fp32).
// ---------------------------------------------------------------------------

typedef float v2f __attribute__((ext_vector_type(2)));
typedef float v8f __attribute__((ext_vector_type(8)));

#define S_LEN 2048
#define DM 1024
#define ROWCHUNKS 32
#define ROWS_PER_CHUNK (S_LEN / ROWCHUNKS) // 64

// --- Kernel A: partial column sums of v --------------------------------------
// grid (4, 32, 2), block 256. Each block sums 64 rows of a 256-column slice.
__global__ void colsum_partial(const float* __restrict__ v,
                               float* __restrict__ partial) {
    int col   = blockIdx.x * blockDim.x + threadIdx.x; // 0..1023
    int chunk = blockIdx.y;                            // 0..31
    int n     = blockIdx.z;                            // 0..1
    const float* base =
        v + ((size_t)n * S_LEN + (size_t)chunk * ROWS_PER_CHUNK) * DM + col;
    float acc = 0.f;
#pragma unroll 8
    for (int r = 0; r < ROWS_PER_CHUNK; ++r)
        acc += base[(size_t)r * DM];
    partial[((size_t)n * ROWCHUNKS + chunk) * DM + col] = acc;
}

// --- Kernel B: reduce partials -> vsum[2][1024] ------------------------------
__global__ void colsum_final(const float* __restrict__ partial,
                             float* __restrict__ vsum) {
    int t = blockIdx.x * blockDim.x + threadIdx.x; // 0..2047
    int n = t >> 10;
    int e = t & (DM - 1);
    float acc = 0.f;
#pragma unroll
    for (int c = 0; c < ROWCHUNKS; ++c)
        acc += partial[((size_t)n * ROWCHUNKS + c) * DM + e];
    vsum[t] = acc;
}

// --- Kernel C/D: y[n,f] = sum_e x[n,e]*W[f,e] + biasScale*bias[f] ------------
// One wave (32 threads) per 16-column N-tile; M dim = batch (rows 0..1 live,
// 2..15 zero-padded). K marches 4 fp32 elements per V_WMMA_F32_16X16X4_F32.
// A 16x4 layout (ISA 7.12.2): lane%16 = M; lanes<16 hold K={0,1}, lanes>=16
// hold K={2,3} across the 2 A VGPRs. B 4x16 mirrors it with lane%16 = N.
__global__ void __launch_bounds__(32)
matvec2_wmma(const float* __restrict__ x,    // [2][1024]
             const float* __restrict__ W,    // [1024][1024], row-major [f][e]
             const float* __restrict__ bias, // [1024]
             float biasScale,
             float* __restrict__ y)          // [2][1024]
{
    const int lane = threadIdx.x;       // 0..31
    const int m    = lane & 15;         // A-row / B-column within tile
    const int kb   = (lane >> 4) << 1;  // K sub-offset: 0 or 2
    const int j    = blockIdx.x;        // N tile 0..63
    const int frow = j * 16 + m;        // W row / output feature index

    const bool mvalid = (m < 2);
    // Safe address for all lanes; padded lanes select 0 afterwards.
    const float2* xr2  = (const float2*)(x + (size_t)(mvalid ? m : 0) * DM);
    const float2* Wrow = (const float2*)(W + (size_t)frow * DM);

    v8f c = {};
#pragma unroll 4
    for (int k = 0; k < DM / 4; ++k) {
        const int e2 = (k * 4 + kb) >> 1; // float2 index (always even element)
        float2 av = xr2[e2];
        float2 bw = Wrow[e2];
        v2f a, b;
        a.x = mvalid ? av.x : 0.f;
        a.y = mvalid ? av.y : 0.f;
        b.x = bw.x;
        b.y = bw.y;
        // D = A(16x4, f32) x B(4x16, f32) + C
        c = __builtin_amdgcn_wmma_f32_16x16x4_f32(
                /*neg_a=*/false, a, /*neg_b=*/false, b,
                /*c_mod=*/(short)0, c, /*reuse_a=*/false, /*reuse_b=*/false);
    }

    // D rows 0 (batch 0) and 1 (batch 1) live in c[0], c[1] of lanes 0..15,
    // with N = lane.
    if (lane < 16) {
        const int f  = j * 16 + lane;
        const float bb = biasScale * bias[f];
        y[f]      = c[0] + bb;
        y[DM + f] = c[1] + bb;
    }
}

// --- Kernel E: broadcast O[n,:] to every sequence position -------------------
// 1M float4 stores, fully coalesced; O rows stay resident in L2.
__global__ void broadcast_out(const float4* __restrict__ O4,
                              float4* __restrict__ out4) {
    size_t i  = (size_t)blockIdx.x * blockDim.x + threadIdx.x; // 0..1048575
    int row   = (int)(i >> 8);  // 256 float4 per 1024-float feature row
    int fq    = (int)(i & 255);
    int n     = row >> 11;      // 2048 rows per batch
    out4[i]   = O4[n * 256 + fq];
}

extern "C" void kernel_launch(void* const* d_in, const int* in_sizes, int n_in,
                              void* d_out, int out_size, void* d_ws, size_t ws_size,
                              hipStream_t stream) {
    (void)in_sizes; (void)n_in; (void)out_size; (void)ws_size;
    // setup_inputs order: q,k,v,mask,Wq,bq,Wk,bk,Wv,bv,Wo,bo
    const float* v  = (const float*)d_in[2];
    const float* Wv = (const float*)d_in[8];
    const float* bv = (const float*)d_in[9];
    const float* Wo = (const float*)d_in[10];
    const float* bo = (const float*)d_in[11];
    float* out = (float*)d_out;

    float* ws      = (float*)d_ws;
    float* partial = ws;                  // 2*32*1024 = 65536 floats
    float* vsum    = ws + 65536;          // 2048 floats
    float* Vsum2   = ws + 65536 + 2048;   // 2048 floats
    float* Ovec    = ws + 65536 + 4096;   // 2048 floats
    // Every ws word is written before it is read; no init needed, no atomics.

    colsum_partial<<<dim3(4, 32, 2), 256, 0, stream>>>(v, partial);
    colsum_final<<<dim3(8), 256, 0, stream>>>(partial, vsum);
    matvec2_wmma<<<dim3(64), 32, 0, stream>>>(vsum, Wv, bv, (float)S_LEN, Vsum2);
    matvec2_wmma<<<dim3(64), 32, 0, stream>>>(Vsum2, Wo, bo, 1.0f, Ovec);
    broadcast_out<<<dim3(4096), 256, 0, stream>>>((const float4*)Ovec,
                                                  (float4*)out);
}